// MlpAttention_7524782702658
// MI455X (gfx1250) — compile-verified
//
#include <hip/hip_runtime.h>
#include <hip/hip_bf16.h>
#include <math.h>

// Problem constants (match reference)
#define BB 32
#define SS 2048
#define QQ 1024
#define HH 1024
#define VV 1024

typedef float v2f __attribute__((ext_vector_type(2)));
typedef float v8f __attribute__((ext_vector_type(8)));

// ---------------------------------------------------------------------------
// Fast tanh: gfx1250 has v_tanh_f32 (1 TRANS op). Fall back to exp-based.
// ---------------------------------------------------------------------------
__device__ __forceinline__ float fast_tanh(float x) {
#if __has_builtin(__builtin_amdgcn_tanhf)
    return __builtin_amdgcn_tanhf(x);
#elif __has_builtin(__builtin_amdgcn_tanh_f32)
    return __builtin_amdgcn_tanh_f32(x);
#else
    float e = __expf(2.0f * x);           // v_exp_f32
    return 1.0f - 2.0f / (e + 1.0f);      // v_rcp_f32 path
#endif
}

// ---------------------------------------------------------------------------
// K0: zero the context accumulator region of d_out
// ---------------------------------------------------------------------------
__global__ __launch_bounds__(256) void zero_kernel(float* __restrict__ p, int n) {
    int i = blockIdx.x * 256 + threadIdx.x;
    if (i < n) p[i] = 0.0f;
}

// ---------------------------------------------------------------------------
// K1: projected_query[b,h] = sum_q query[b,q] * W_query[q,h]
//     M=32 (B), N=1024 (H), K=1024 (Q) via V_WMMA_F32_16X16X4_F32.
//     One wave per 16x16 output tile; grid = (64 Ntiles, 2 Mtiles).
//
//     A (16x4 f32): lanes 0-15 hold M=lane, {K, K+1} in v0/v1;
//                   lanes 16-31 hold M=lane-16, {K+2, K+3}.
//     B (4x16 f32): lanes hold N=lane%16; v0/v1 hold two K rows,
//                   lane-half selects K offset 0 or 2 (mirror of A).
//     C/D (16x16 f32): lane n = lane%16, VGPR r -> M = r + (lane>=16)*8.
// ---------------------------------------------------------------------------
__global__ __launch_bounds__(32) void pq_wmma_kernel(const float* __restrict__ query,
                                                     const float* __restrict__ Wq,
                                                     float* __restrict__ pq) {
    const int nBase = blockIdx.x * 16;
    const int mBase = blockIdx.y * 16;
    const int lane  = threadIdx.x;        // 0..31, full wave (EXEC all 1s)
    const int half  = lane >> 4;          // 0 or 1
    const int l16   = lane & 15;

    v8f c = {};
    for (int k = 0; k < QQ; k += 4) {
        const int ka = k + half * 2;
        v2f a, b;
        a.x = query[(size_t)(mBase + l16) * QQ + ka];
        a.y = query[(size_t)(mBase + l16) * QQ + ka + 1];
        b.x = Wq[(size_t)(ka)     * HH + nBase + l16];
        b.y = Wq[(size_t)(ka + 1) * HH + nBase + l16];
        // 8 args: (neg_a, A, neg_b, B, c_mod, C, reuse_a, reuse_b)
        c = __builtin_amdgcn_wmma_f32_16x16x4_f32(false, a, false, b,
                                                  (short)0, c, false, false);
    }

#pragma unroll
    for (int r = 0; r < 8; ++r) {
        const int m = mBase + r + half * 8;
        const int n = nBase + l16;
        pq[(size_t)m * HH + n] = c[r];
    }
}

// ---------------------------------------------------------------------------
// K2: raw_scores[b,s] = sum_h tanh(keys[b,s,h] + pq[b,h]) * We[h]   (+mask)
//     One wave per (b,s) row; 8 waves/block; float4 streaming loads.
//     Memory-bound: 256 MB of keys, ~roofline with hw tanh.
// ---------------------------------------------------------------------------
__global__ __launch_bounds__(256) void energy_scores_kernel(
        const float* __restrict__ keys, const float* __restrict__ pq,
        const float* __restrict__ We, const unsigned char* __restrict__ mask,
        float* __restrict__ scores_raw) {
    __shared__ float sWe[HH];
    for (int i = threadIdx.x; i < HH; i += 256) sWe[i] = We[i];
    __syncthreads();

    const int wave = threadIdx.x >> 5;
    const int lane = threadIdx.x & 31;
    const size_t row = (size_t)blockIdx.x * 8 + wave;   // b*S + s
    const int b = (int)(row >> 11);                     // row / 2048

    const float4* krow = reinterpret_cast<const float4*>(keys + (row << 10));
    const float4* prow = reinterpret_cast<const float4*>(pq + ((size_t)b << 10));
    const float4* wrow = reinterpret_cast<const float4*>(sWe);

    float acc = 0.0f;
#pragma unroll
    for (int i = 0; i < 8; ++i) {
        const int idx = lane + i * 32;                  // coalesced float4
        const float4 kv = krow[idx];
        const float4 pv = prow[idx];
        const float4 wv = wrow[idx];
        acc += fast_tanh(kv.x + pv.x) * wv.x;
        acc += fast_tanh(kv.y + pv.y) * wv.y;
        acc += fast_tanh(kv.z + pv.z) * wv.z;
        acc += fast_tanh(kv.w + pv.w) * wv.w;
    }
#pragma unroll
    for (int off = 16; off > 0; off >>= 1)
        acc += __shfl_xor(acc, off, 32);

    if (lane == 0) {
        scores_raw[row] = mask[row] ? acc : -INFINITY;
    }
}

// ---------------------------------------------------------------------------
// K3: softmax over S=2048 per batch, in place. 256 threads, 8 vals/thread.
// ---------------------------------------------------------------------------
__global__ __launch_bounds__(256) void softmax_kernel(float* __restrict__ scores) {
    float* p = scores + ((size_t)blockIdx.x << 11);
    const int t    = threadIdx.x;
    const int wave = t >> 5;
    const int lane = t & 31;
    __shared__ float smem[8];

    float v[8];
    float m = -INFINITY;
#pragma unroll
    for (int i = 0; i < 8; ++i) { v[i] = p[t + (i << 8)]; m = fmaxf(m, v[i]); }
#pragma unroll
    for (int off = 16; off > 0; off >>= 1) m = fmaxf(m, __shfl_xor(m, off, 32));
    if (lane == 0) smem[wave] = m;
    __syncthreads();
    float bm = smem[0];
#pragma unroll
    for (int w = 1; w < 8; ++w) bm = fmaxf(bm, smem[w]);
    __syncthreads();

    float sum = 0.0f;
#pragma unroll
    for (int i = 0; i < 8; ++i) { v[i] = __expf(v[i] - bm); sum += v[i]; }
#pragma unroll
    for (int off = 16; off > 0; off >>= 1) sum += __shfl_xor(sum, off, 32);
    if (lane == 0) smem[wave] = sum;
    __syncthreads();
    float bs = 0.0f;
#pragma unroll
    for (int w = 0; w < 8; ++w) bs += smem[w];

    const float inv = 1.0f / bs;
#pragma unroll
    for (int i = 0; i < 8; ++i) p[t + (i << 8)] = v[i] * inv;
}

// ---------------------------------------------------------------------------
// K4: context[b,v] += sum_{s in chunk} probs[b,s] * values[b,s,v]
//     grid = (16 s-chunks, 32 batches); 256 threads x float4 cover V=1024.
//     global_atomic_add_f32 accumulation; 512 blocks to saturate HBM.
// ---------------------------------------------------------------------------
__global__ __launch_bounds__(256) void context_kernel(
        const float* __restrict__ values, const float* __restrict__ probs,
        float* __restrict__ ctx) {
    const int chunk = blockIdx.x;   // 0..15 (128 s each)
    const int b     = blockIdx.y;   // 0..31

    __shared__ float sp[128];
    if (threadIdx.x < 128)
        sp[threadIdx.x] = probs[((size_t)b << 11) + chunk * 128 + threadIdx.x];
    __syncthreads();

    const float4* vb = reinterpret_cast<const float4*>(
        values + ((size_t)b * SS + (size_t)chunk * 128) * VV);
    const int col4 = threadIdx.x;   // 256 * 4 = 1024 columns

    float4 acc = make_float4(0.f, 0.f, 0.f, 0.f);
    for (int s = 0; s < 128; ++s) {
        const float w = sp[s];
        const float4 vv = vb[(size_t)s * 256 + col4];
        acc.x += w * vv.x; acc.y += w * vv.y;
        acc.z += w * vv.z; acc.w += w * vv.w;
    }

    float* out = ctx + (size_t)b * VV + col4 * 4;
    atomicAdd(out + 0, acc.x);
    atomicAdd(out + 1, acc.y);
    atomicAdd(out + 2, acc.z);
    atomicAdd(out + 3, acc.w);
}

// ---------------------------------------------------------------------------
// Launch: d_in order = {query, projected_keys, values, mask, W_query, W_energy}
// d_out = [context (32*1024) | scores (32*2048)] fp32.
// d_ws  = pq scratch (32*1024 floats = 128 KB).
// ---------------------------------------------------------------------------
extern "C" void kernel_launch(void* const* d_in, const int* in_sizes, int n_in,
                              void* d_out, int out_size, void* d_ws, size_t ws_size,
                              hipStream_t stream) {
    const float*         query  = (const float*)d_in[0];
    const float*         keys   = (const float*)d_in[1];
    const float*         values = (const float*)d_in[2];
    const unsigned char* mask   = (const unsigned char*)d_in[3];
    const float*         Wq     = (const float*)d_in[4];
    const float*         We     = (const float*)d_in[5];

    float* out    = (float*)d_out;
    float* ctx    = out;                 // [B, V]
    float* scores = out + BB * VV;       // [B, S]
    float* pq     = (float*)d_ws;        // [B, H]

    zero_kernel<<<(BB * VV + 255) / 256, 256, 0, stream>>>(ctx, BB * VV);
    pq_wmma_kernel<<<dim3(HH / 16, BB / 16), 32, 0, stream>>>(query, Wq, pq);
    energy_scores_kernel<<<(BB * SS) / 8, 256, 0, stream>>>(keys, pq, We, mask, scores);
    softmax_kernel<<<BB, 256, 0, stream>>>(scores);
    context_kernel<<<dim3(SS / 128, BB), 256, 0, stream>>>(values, scores, ctx);
}